// BatchedGAT_37263136260739
// MI455X (gfx1250) — compile-verified
//
#include <hip/hip_runtime.h>
#include <hip/hip_bf16.h>

typedef __attribute__((ext_vector_type(2))) float v2f;
typedef __attribute__((ext_vector_type(8))) float v8f;

#define BATCH    8
#define NNODES   512
#define IN_DIM   256
#define HEADS    4
#define PER_HEAD 32
#define OUT_DIM  128
#define BN       (BATCH * NNODES)   // 4096 rows

// ---------------------------------------------------------------------------
// CDNA5 async global->LDS copy (16B per lane), tracked with ASYNCcnt.
// VDST operand = LDS byte offset (low 32 bits of the generic pointer: the LDS
// aperture is keyed on addr[63:32], addr[31:0] is the in-wave LDS offset).
// ---------------------------------------------------------------------------
__device__ __forceinline__ void async_copy_b128(void* lds_dst, const void* gsrc) {
  unsigned lds_off = (unsigned)(unsigned long long)lds_dst;
  unsigned long long gaddr = (unsigned long long)gsrc;
  asm volatile("global_load_async_to_lds_b128 %0, %1, off"
               :: "v"(lds_off), "v"(gaddr) : "memory");
}

__device__ __forceinline__ void async_wait0() {
  asm volatile("s_wait_asynccnt 0x0" ::: "memory");
}

// ---------------------------------------------------------------------------
// Stage A: projection GEMM  out[4096,128] = x[4096,256] @ W[256,128] + b
// One block = one 16-row tile; 8 waves each own a 16-wide column tile.
// f32 WMMA 16x16x4, K loop over 256.
// ---------------------------------------------------------------------------
__global__ __launch_bounds__(256) void gat_proj_kernel(
    const float* __restrict__ x, const float* __restrict__ W,
    const float* __restrict__ bvec, float* __restrict__ out) {
  // +4 pad: 16B-aligned rows for async B128, bank stride 4 mod 64 -> the
  // 16-lane column read xs[m][k] still hits 16 distinct banks.
  __shared__ float xs[16][IN_DIM + 4];

  const int m0 = blockIdx.x * 16;
  const int t  = threadIdx.x;

  // Async-stage the x tile: 16 rows x 256 floats = 1024 x 16B chunks.
  for (int idx = t; idx < 16 * 64; idx += 256) {
    int r = idx >> 6;
    int q = idx & 63;
    async_copy_b128(&xs[r][q * 4], &x[(m0 + r) * IN_DIM + q * 4]);
  }
  async_wait0();
  __syncthreads();

  const int wave = t >> 5;
  const int lane = t & 31;
  const int n0   = wave * 16;
  const int m    = lane & 15;
  const int kb   = (lane < 16) ? 0 : 2;   // A/B K sub-offset per ISA layout
  const int n    = lane & 15;

  v8f acc = {};
  for (int k0 = 0; k0 < IN_DIM; k0 += 4) {
    v2f a, bf;
    a.x  = xs[m][k0 + kb];
    a.y  = xs[m][k0 + kb + 1];
    bf.x = W[(k0 + kb) * OUT_DIM + n0 + n];
    bf.y = W[(k0 + kb + 1) * OUT_DIM + n0 + n];
    acc = __builtin_amdgcn_wmma_f32_16x16x4_f32(
        false, a, false, bf, (short)0, acc, false, false);
  }

  const float bb = bvec[n0 + n];
  const int mbase = (lane < 16) ? 0 : 8;
#pragma unroll
  for (int r = 0; r < 8; ++r) {
    out[(m0 + mbase + r) * OUT_DIM + n0 + n] = acc[r] + bb;
  }
}

// ---------------------------------------------------------------------------
// Stage B: fused logits + masked softmax + WMMA aggregation.
// One block = (batch b, 16-target tile i0). 256 threads = 8 waves.
// Per head: e_lds[16][512] logits in LDS, row softmax, then
// alpha[16x512] @ xl[512x32] via f32 WMMA split over 8 waves.
// ---------------------------------------------------------------------------
__global__ __launch_bounds__(256) void gat_attn_kernel(
    const float* __restrict__ xl, const float* __restrict__ xr,
    const int* __restrict__ adj, const float* __restrict__ att,
    const float* __restrict__ bias, float* __restrict__ out) {
  __shared__ float e_lds[16][NNODES + 1];   // 32.8 KB, padded stride 513
  __shared__ float xl_s[128][PER_HEAD];     // 16 KB source-feature chunk
  __shared__ float xr_s[16][PER_HEAD];      // 2 KB target-feature tile
  __shared__ float att_s[PER_HEAD];
  __shared__ float part[8][8][32];          // 8 KB WMMA partials

  const int b  = blockIdx.y;
  const int i0 = blockIdx.x * 16;
  const int t  = threadIdx.x;

  const float* xl_b  = xl + (size_t)b * NNODES * OUT_DIM;
  const float* xr_b  = xr + (size_t)b * NNODES * OUT_DIM;
  const int*   adj_b = adj + (size_t)b * NNODES * NNODES;

  const int wave = t >> 5;
  const int lane = t & 31;

  for (int h = 0; h < HEADS; ++h) {
    const int hc0 = h * PER_HEAD;

    // ---- async-stage xr tile (16x32 = 128 x 16B) + attention vector ----
    if (t < 128) {
      async_copy_b128(&xr_s[t >> 3][(t & 7) * 4],
                      &xr_b[(i0 + (t >> 3)) * OUT_DIM + hc0 + (t & 7) * 4]);
    }
    if (t < PER_HEAD) att_s[t] = att[hc0 + t];
    async_wait0();
    __syncthreads();

    // ---- Phase 1: logits e[i][j] with adjacency mask (diag forced) ----
    {
      const int i    = t & 15;
      const int jgrp = t >> 4;               // 0..15
      for (int jc = 0; jc < NNODES; jc += 128) {
        // async-stage xl chunk: 128 rows x 32 floats = 1024 x 16B chunks
        for (int idx = t; idx < 128 * 8; idx += 256) {
          int jj = idx >> 3, q = idx & 7;
          async_copy_b128(&xl_s[jj][q * 4],
                          &xl_b[(jc + jj) * OUT_DIM + hc0 + q * 4]);
        }
        async_wait0();
        __syncthreads();
#pragma unroll
        for (int q = 0; q < 8; ++q) {
          const int jj = jgrp * 8 + q;
          const int j  = jc + jj;
          float acc = 0.f;
#pragma unroll
          for (int c = 0; c < PER_HEAD; ++c) {
            float s = xr_s[i][c] + xl_s[jj][c];
            s = fmaxf(s, 0.2f * s);          // LeakyReLU(0.2), slope < 1
            acc = fmaf(s, att_s[c], acc);
          }
          const bool conn = (adj_b[j * NNODES + (i0 + i)] != 0) || (j == i0 + i);
          e_lds[i][j] = conn ? acc : -1e30f;
        }
        __syncthreads();
      }
    }

    // ---- Phase 2: row softmax over j (16 lanes per row, shfl reduce) ----
    {
      const int row = t >> 4;
      const int sub = t & 15;
      float mx = -1e30f;
      for (int j = sub; j < NNODES; j += 16) mx = fmaxf(mx, e_lds[row][j]);
#pragma unroll
      for (int off = 8; off; off >>= 1) mx = fmaxf(mx, __shfl_xor(mx, off, 32));
      float sum = 0.f;
      for (int j = sub; j < NNODES; j += 16) {
        float v = __expf(e_lds[row][j] - mx);   // masked -> exp underflows to 0
        e_lds[row][j] = v;
        sum += v;
      }
#pragma unroll
      for (int off = 8; off; off >>= 1) sum += __shfl_xor(sum, off, 32);
      const float inv = 1.0f / sum;
      for (int j = sub; j < NNODES; j += 16) e_lds[row][j] *= inv;
    }
    __syncthreads();

    // ---- Phase 3: out_tile = alpha[16x512] @ xl[512x32] via WMMA ----
    {
      const int ct = wave & 1;                // column tile (0..1) of 16
      const int js = (wave >> 1) * 128;       // K segment of 128
      const int m  = lane & 15;
      const int kb = (lane < 16) ? 0 : 2;
      const int n  = lane & 15;
      const int c0 = hc0 + ct * 16;

      v8f acc = {};
      for (int k = 0; k < 32; ++k) {
        const int j0 = js + k * 4;
        v2f a, bf;
        a.x  = e_lds[m][j0 + kb];
        a.y  = e_lds[m][j0 + kb + 1];
        bf.x = xl_b[(j0 + kb) * OUT_DIM + c0 + n];
        bf.y = xl_b[(j0 + kb + 1) * OUT_DIM + c0 + n];
        acc = __builtin_amdgcn_wmma_f32_16x16x4_f32(
            false, a, false, bf, (short)0, acc, false, false);
      }
#pragma unroll
      for (int r = 0; r < 8; ++r) part[wave][r][lane] = acc[r];
    }
    __syncthreads();

    // ---- reduce 4 K-segment partials and store with bias ----
    for (int e0 = t; e0 < 16 * PER_HEAD; e0 += 256) {
      const int ii  = e0 >> 5;
      const int c   = e0 & 31;
      const int ctt = c >> 4;
      const int nn  = c & 15;
      const int r   = ii & 7;
      const int lA  = nn + ((ii >= 8) ? 16 : 0);
      const float v = part[ctt][r][lA] + part[ctt + 2][r][lA] +
                      part[ctt + 4][r][lA] + part[ctt + 6][r][lA];
      out[((size_t)b * NNODES + i0 + ii) * OUT_DIM + hc0 + c] = v + bias[hc0 + c];
    }
    __syncthreads();   // e_lds / xl_s reused next head
  }
}

// ---------------------------------------------------------------------------
extern "C" void kernel_launch(void* const* d_in, const int* in_sizes, int n_in,
                              void* d_out, int out_size, void* d_ws, size_t ws_size,
                              hipStream_t stream) {
  const float* x    = (const float*)d_in[0];
  const int*   adj  = (const int*)d_in[1];
  const float* W_l  = (const float*)d_in[2];
  const float* b_l  = (const float*)d_in[3];
  const float* W_r  = (const float*)d_in[4];
  const float* b_r  = (const float*)d_in[5];
  const float* att  = (const float*)d_in[6];
  const float* bias = (const float*)d_in[7];
  float* out = (float*)d_out;

  float* xl = (float*)d_ws;                       // [4096,128]
  float* xr = xl + (size_t)BN * OUT_DIM;          // [4096,128]

  gat_proj_kernel<<<dim3(BN / 16), 256, 0, stream>>>(x, W_l, b_l, xl);
  gat_proj_kernel<<<dim3(BN / 16), 256, 0, stream>>>(x, W_r, b_r, xr);
  gat_attn_kernel<<<dim3(NNODES / 16, BATCH), 256, 0, stream>>>(
      xl, xr, adj, att, bias, out);
}